// CoAttention_33054068310066
// MI455X (gfx1250) — compile-verified
//
#include <hip/hip_runtime.h>

typedef __attribute__((ext_vector_type(2))) float v2f;
typedef __attribute__((ext_vector_type(8))) float v8f;

#define NEGV (-1e30f)
constexpr int Bn = 32, Dd = 128, CL = 1024, QL = 256;

// ---------------- K0: row/col bias vectors -------------------------------
__global__ void k_bias(const float* __restrict__ C, const float* __restrict__ Q,
                       const float* __restrict__ W0,
                       float* __restrict__ rowb, float* __restrict__ colb) {
  int t = blockIdx.x * blockDim.x + threadIdx.x;
  if (t < Bn * CL) {
    int b = t / CL, i = t % CL;
    float s = 0.f;
    for (int k = 0; k < Dd; ++k) s += C[(size_t)(b * Dd + k) * CL + i] * W0[k];
    rowb[t] = s;
  } else if (t < Bn * CL + Bn * QL) {
    int u = t - Bn * CL;
    int b = u / QL, j = u % QL;
    float s = 0.f;
    for (int k = 0; k < Dd; ++k) s += Q[(size_t)(b * Dd + k) * QL + j] * W0[Dd + k];
    colb[u] = s;
  }
}

// ---------------- K_t: LDS-tiled transpose C(B,D,CL) -> Ct(B,CL,D) -------
__global__ void k_transpose(const float* __restrict__ C, float* __restrict__ Ct) {
  __shared__ float tile[32][33];
  int b = blockIdx.z;
  int i0 = blockIdx.x * 32;   // along CL
  int k0 = blockIdx.y * 32;   // along D
  int tx = threadIdx.x, ty = threadIdx.y;  // 32 x 8
  for (int r = 0; r < 32; r += 8)
    tile[ty + r][tx] = C[(size_t)(b * Dd + k0 + ty + r) * CL + i0 + tx];
  __syncthreads();
  for (int r = 0; r < 32; r += 8)
    Ct[(size_t)(b * CL + i0 + ty + r) * Dd + k0 + tx] = tile[tx][ty + r];
}

// ---------------- K1: S = (w3*C_t) . Q_t^T  (WMMA f32 16x16x4) ----------
__global__ void k_sgemm(const float* __restrict__ C, const float* __restrict__ Q,
                        const float* __restrict__ W0, float* __restrict__ S) {
  const int lane = threadIdx.x & 31;
  const int wave = blockIdx.x * (blockDim.x >> 5) + (threadIdx.x >> 5);
  const int nN = QL / 16, nM = CL / 16;
  int b = wave / (nM * nN);
  int rem = wave % (nM * nN);
  int i0 = (rem / nN) * 16;
  int j0 = (rem % nN) * 16;
  const float* w3 = W0 + 2 * Dd;
  int half = lane >> 4, mn = lane & 15;
  v8f acc = {};
  for (int k0 = 0; k0 < Dd; k0 += 4) {
    int kk = k0 + half * 2;
    v2f a, bq;
    a.x  = w3[kk]     * C[(size_t)(b * Dd + kk)     * CL + i0 + mn];
    a.y  = w3[kk + 1] * C[(size_t)(b * Dd + kk + 1) * CL + i0 + mn];
    bq.x = Q[(size_t)(b * Dd + kk)     * QL + j0 + mn];
    bq.y = Q[(size_t)(b * Dd + kk + 1) * QL + j0 + mn];
    acc = __builtin_amdgcn_wmma_f32_16x16x4_f32(false, a, false, bq,
                                                (short)0, acc, false, false);
  }
  float* dst = S + (size_t)(b * CL + i0 + half * 8) * QL + j0 + mn;
#pragma unroll
  for (int v = 0; v < 8; ++v) dst[(size_t)v * QL] = acc[v];
}

// ---------------- K2: softmax over j (len 256), one wave per row ---------
__global__ void k_softmax_row(const float* __restrict__ S, const float* __restrict__ colb,
                              const float* __restrict__ qm, float* __restrict__ Sbar) {
  int lane = threadIdx.x & 31;
  int wid = threadIdx.x >> 5;
  int row = blockIdx.x * 8 + wid;  // over B*CL
  int b = row / CL;
  const float* src = S + (size_t)row * QL;
  float vals[8];
  float m = NEGV;
#pragma unroll
  for (int t = 0; t < 8; ++t) {
    int j = lane + 32 * t;
    float q = qm[b * QL + j];
    float v = (src[j] + colb[b * QL + j]) * q + NEGV * (1.0f - q);
    vals[t] = v;
    m = fmaxf(m, v);
  }
#pragma unroll
  for (int off = 16; off > 0; off >>= 1) m = fmaxf(m, __shfl_xor(m, off, 32));
  float s = 0.f;
#pragma unroll
  for (int t = 0; t < 8; ++t) { vals[t] = __expf(vals[t] - m); s += vals[t]; }
#pragma unroll
  for (int off = 16; off > 0; off >>= 1) s += __shfl_xor(s, off, 32);
  float inv = 1.0f / s;
  float* dst = Sbar + (size_t)row * QL;
#pragma unroll
  for (int t = 0; t < 8; ++t) dst[lane + 32 * t] = vals[t] * inv;
}

// ---------------- K3: softmax over i (len 1024), in place ---------------
__global__ void k_softmax_col(float* __restrict__ S, const float* __restrict__ rowb,
                              const float* __restrict__ cm) {
  __shared__ float sm[4][64], ss[4][64], fm[64], fs[64];
  int b = blockIdx.x / (QL / 64);
  int j = (blockIdx.x % (QL / 64)) * 64 + (threadIdx.x & 63);
  int jj = threadIdx.x & 63;
  int isub = threadIdx.x >> 6;  // 0..3
  float m = NEGV, s = 0.f;
  for (int i = isub; i < CL; i += 4) {
    float c = cm[b * CL + i];
    float v = (S[(size_t)(b * CL + i) * QL + j] + rowb[b * CL + i]) * c + NEGV * (1.0f - c);
    float nm = fmaxf(m, v);
    s = s * __expf(m - nm) + __expf(v - nm);
    m = nm;
  }
  sm[isub][jj] = m; ss[isub][jj] = s;
  __syncthreads();
  if (isub == 0) {
    float M = sm[0][jj], Su = ss[0][jj];
#pragma unroll
    for (int r = 1; r < 4; ++r) {
      float nm = fmaxf(M, sm[r][jj]);
      Su = Su * __expf(M - nm) + ss[r][jj] * __expf(sm[r][jj] - nm);
      M = nm;
    }
    fm[jj] = M; fs[jj] = 1.0f / Su;
  }
  __syncthreads();
  float M = fm[jj], inv = fs[jj];
  for (int i = isub; i < CL; i += 4) {
    float c = cm[b * CL + i];
    size_t idx = (size_t)(b * CL + i) * QL + j;
    float v = (S[idx] + rowb[b * CL + i]) * c + NEGV * (1.0f - c);
    S[idx] = __expf(v - M) * inv;
  }
}

// ---------------- K4: T = S_bbar^T . C_t  (256x128, K=1024) -------------
__global__ void k_tgemm(const float* __restrict__ Sbb, const float* __restrict__ Ct,
                        float* __restrict__ T) {
  int lane = threadIdx.x & 31;
  int wave = blockIdx.x * (blockDim.x >> 5) + (threadIdx.x >> 5);
  int b = wave / (16 * 8);
  int rem = wave % (16 * 8);
  int j0 = (rem / 8) * 16;
  int k0 = (rem % 8) * 16;
  int half = lane >> 4, mn = lane & 15;
  v8f acc = {};
  for (int i0 = 0; i0 < CL; i0 += 4) {
    int ii = i0 + half * 2;
    v2f a, bb;
    a.x  = Sbb[(size_t)(b * CL + ii)     * QL + j0 + mn];
    a.y  = Sbb[(size_t)(b * CL + ii + 1) * QL + j0 + mn];
    bb.x = Ct[(size_t)(b * CL + ii)      * Dd + k0 + mn];
    bb.y = Ct[(size_t)(b * CL + ii + 1)  * Dd + k0 + mn];
    acc = __builtin_amdgcn_wmma_f32_16x16x4_f32(false, a, false, bb,
                                                (short)0, acc, false, false);
  }
  float* dst = T + (size_t)(b * QL + j0 + half * 8) * Dd + k0 + mn;
#pragma unroll
  for (int v = 0; v < 8; ++v) dst[(size_t)v * Dd] = acc[v];
}

// ---------------- K5: A = Sbar.Q_t ; Bm = Sbar.T (fused, shared A-op) ----
__global__ void k_abgemm(const float* __restrict__ Sbar, const float* __restrict__ Q,
                         const float* __restrict__ T,
                         float* __restrict__ Am, float* __restrict__ Bm) {
  int lane = threadIdx.x & 31;
  int wave = blockIdx.x * (blockDim.x >> 5) + (threadIdx.x >> 5);
  int b = wave / (64 * 8);
  int rem = wave % (64 * 8);
  int i0 = (rem / 8) * 16;
  int k0 = (rem % 8) * 16;
  int half = lane >> 4, mn = lane & 15;
  v8f accA = {}, accB = {};
  for (int j0 = 0; j0 < QL; j0 += 4) {
    int jj = j0 + half * 2;
    v2f a, bq, bt;
    a.x  = Sbar[(size_t)(b * CL + i0 + mn) * QL + jj];
    a.y  = Sbar[(size_t)(b * CL + i0 + mn) * QL + jj + 1];
    bq.x = Q[(size_t)(b * Dd + k0 + mn) * QL + jj];
    bq.y = Q[(size_t)(b * Dd + k0 + mn) * QL + jj + 1];
    bt.x = T[(size_t)(b * QL + jj)     * Dd + k0 + mn];
    bt.y = T[(size_t)(b * QL + jj + 1) * Dd + k0 + mn];
    accA = __builtin_amdgcn_wmma_f32_16x16x4_f32(false, a, false, bq,
                                                 (short)0, accA, false, false);
    accB = __builtin_amdgcn_wmma_f32_16x16x4_f32(false, a, false, bt,
                                                 (short)0, accB, false, false);
  }
  int k = k0 + mn;
#pragma unroll
  for (int v = 0; v < 8; ++v) {
    size_t idx = (size_t)(b * CL + i0 + half * 8 + v) * Dd + k;
    Am[idx] = accA[v];
    Bm[idx] = accB[v];
  }
}

// ---------------- K6: fused depthwise(5,pad2) + pointwise conv -----------
__global__ void k_conv(const float* __restrict__ Ct, const float* __restrict__ Am,
                       const float* __restrict__ Bmat,
                       const float* __restrict__ dww, const float* __restrict__ dwb,
                       const float* __restrict__ pww, const float* __restrict__ pwb,
                       float* __restrict__ out) {
  __shared__ float dwX[512][17];  // padded -> bank-conflict free
  int b = blockIdx.x / (CL / 16);
  int p0 = (blockIdx.x % (CL / 16)) * 16;
  int d = threadIdx.x & 127;
  int grp = threadIdx.x >> 7;  // 0 or 1
  // depthwise conv + bias -> LDS (channels = [Ct, A, Ct*A, Ct*Bm])
  for (int it = 0; it < 32; ++it) {
    int combo = it * 2 + grp;   // 0..63 = (which,p)
    int which = combo >> 4;
    int p = combo & 15;
    float acc = dwb[which * Dd + d];
#pragma unroll
    for (int tap = 0; tap < 5; ++tap) {
      int pos = p0 + p + tap - 2;
      float x = 0.f;
      if (pos >= 0 && pos < CL) {
        size_t base = (size_t)(b * CL + pos) * Dd + d;
        float c = Ct[base];
        if (which == 0)      x = c;
        else if (which == 1) x = Am[base];
        else if (which == 2) x = c * Am[base];
        else                 x = c * Bmat[base];
      }
      acc += dww[(which * Dd + d) * 5 + tap] * x;
    }
    dwX[which * Dd + d][p] = acc;
  }
  __syncthreads();
  // pointwise 128x512 GEMM via WMMA; wave w -> output rows [16w,16w+16)
  int lane = threadIdx.x & 31;
  int o0 = (threadIdx.x >> 5) * 16;
  int half = lane >> 4, mn = lane & 15;
  v8f acc = {};
  for (int c0 = 0; c0 < 512; c0 += 4) {
    int cc = c0 + half * 2;
    v2f a, bb;
    a.x  = pww[(size_t)(o0 + mn) * 512 + cc];
    a.y  = pww[(size_t)(o0 + mn) * 512 + cc + 1];
    bb.x = dwX[cc][mn];
    bb.y = dwX[cc + 1][mn];
    acc = __builtin_amdgcn_wmma_f32_16x16x4_f32(false, a, false, bb,
                                                (short)0, acc, false, false);
  }
#pragma unroll
  for (int v = 0; v < 8; ++v) {
    int o = o0 + half * 8 + v;
    out[(size_t)(b * Dd + o) * CL + p0 + mn] = acc[v] + pwb[o];
  }
}

extern "C" void kernel_launch(void* const* d_in, const int* in_sizes, int n_in,
                              void* d_out, int out_size, void* d_ws, size_t ws_size,
                              hipStream_t stream) {
  const float* C   = (const float*)d_in[0];
  const float* Q   = (const float*)d_in[1];
  const float* cm  = (const float*)d_in[2];
  const float* qm  = (const float*)d_in[3];
  const float* W0  = (const float*)d_in[4];
  const float* dww = (const float*)d_in[5];
  const float* dwb = (const float*)d_in[6];
  const float* pww = (const float*)d_in[7];
  const float* pwb = (const float*)d_in[8];
  float* out = (float*)d_out;

  // workspace layout (floats)
  float* ws = (float*)d_ws;
  const size_t nS = (size_t)Bn * CL * QL;       // 8,388,608
  float* S    = ws;                              // S -> later S_bbar (in place)
  float* Sbar = S + nS;
  float* rowb = Sbar + nS;                       // B*CL
  float* colb = rowb + (size_t)Bn * CL;          // B*QL
  float* T    = colb + (size_t)Bn * QL;          // B*QL*D
  float* Ct   = T + (size_t)Bn * QL * Dd;        // B*CL*D
  float* Am   = Ct + (size_t)Bn * CL * Dd;       // B*CL*D
  float* Bm   = Am + (size_t)Bn * CL * Dd;       // B*CL*D
  (void)ws_size; (void)in_sizes; (void)n_in; (void)out_size;

  k_bias<<<(Bn * CL + Bn * QL + 255) / 256, 256, 0, stream>>>(C, Q, W0, rowb, colb);
  k_transpose<<<dim3(CL / 32, Dd / 32, Bn), dim3(32, 8), 0, stream>>>(C, Ct);
  k_sgemm<<<(Bn * (CL / 16) * (QL / 16)) / 8, 256, 0, stream>>>(C, Q, W0, S);
  k_softmax_row<<<(Bn * CL) / 8, 256, 0, stream>>>(S, colb, qm, Sbar);
  k_softmax_col<<<Bn * (QL / 64), 256, 0, stream>>>(S, rowb, cm);  // S -> S_bbar
  k_tgemm<<<(Bn * (QL / 16) * (Dd / 16)) / 8, 256, 0, stream>>>(S, Ct, T);
  k_abgemm<<<(Bn * (CL / 16) * (Dd / 16)) / 8, 256, 0, stream>>>(Sbar, Q, T, Am, Bm);
  k_conv<<<Bn * (CL / 16), 256, 0, stream>>>(Ct, Am, Bm, dww, dwb, pww, pwb, out);
}